// SingleStageDetector_38208029065866
// MI455X (gfx1250) — compile-verified
//
#include <hip/hip_runtime.h>
#include <hip/hip_bf16.h>

typedef __attribute__((ext_vector_type(2))) float v2f;
typedef __attribute__((ext_vector_type(8))) float v8f;
typedef __attribute__((ext_vector_type(4))) unsigned int v4u;
typedef __attribute__((ext_vector_type(4))) int v4i;
typedef __attribute__((ext_vector_type(8))) int v8i;

#define B_    256
#define CIN_  1280
#define HW_   49
#define HID_  128
#define A_    9
#define NCLS_ 20
#define NGT_  40
#define OUTD_ 65
#define M_TOT (B_ * HW_)          // 12544 = 784 * 16
#define K1STEPS (CIN_ / 4)        // 320
#define K2STEPS (HID_ / 4)        // 32
#define NT1   8                   // 128 / 16
#define NT2   5                   // 80 / 16 (65 padded to 80)

#define CH_   16                  // K-steps per LDS-staged W1 chunk
#define NCHUNK (K1STEPS / CH_)    // 20
#define CHUNK_FLOATS (NT1 * CH_ * 64)   // 8192 floats = 32 KB
#define CHUNK_BYTES  (CHUNK_FLOATS * 4)

// output layout (flat, concatenated in reference return order)
#define OFF_CONF 0
#define OFF_OFFS (B_ * A_ * HW_)                       // 112896
#define OFF_CLS  (OFF_OFFS + B_ * A_ * 4 * HW_)        // 564480
#define OFF_IOU  (OFF_CLS + B_ * NCLS_ * HW_)          // 815360

#define PACKW1_FLOATS (NT1 * K1STEPS * 64)             // 163840
#define PACKW2_FLOATS (NT2 * K2STEPS * 64)             // 10240

#if defined(__has_builtin)
#  if __has_builtin(__builtin_amdgcn_tensor_load_to_lds)
#    define HAVE_TDM 1
#  else
#    define HAVE_TDM 0
#  endif
#else
#  define HAVE_TDM 0
#endif

__device__ __forceinline__ float sigmoid_f(float x) {
    return 1.0f / (1.0f + __expf(-x));
}

#if HAVE_TDM
// Issue a TDM 2-D tile load: 8 rows (nt) x 1024 elements (CH_*64 f32) of packed
// W1 fragments, row stride 20480 elements, into LDS at lds_off. D# layout per
// CDNA5 ISA ch.8: group0 = {count=1, lds_addr, global_addr[56:0], type=2},
// group1 = {data_size=4B, tensor_dim0/1, tile_dim0/1, dim0_stride}.
__device__ __forceinline__ void tdm_load_w1_chunk(const float* packW1, int chunk,
                                                  unsigned lds_off) {
    unsigned long long ga =
        (unsigned long long)(uintptr_t)packW1 + (unsigned long long)chunk * (CH_ * 64 * 4);
    v4u g0 = {};
    g0.x = 1u;                                            // count=1 (user descriptor)
    g0.y = lds_off;                                       // LDS byte address
    g0.z = (unsigned)(ga & 0xFFFFFFFFull);                // global_addr[31:0]
    g0.w = (unsigned)((ga >> 32) & 0x1FFFFFFull) | 0x80000000u;  // addr[56:32] | type=2
    v8i g1 = {};
    g1[0] = 0x00020000;                     // workgroup_mask=0, data_size=2 (4 bytes)
    g1[1] = (CH_ * 64) << 16;               // tensor_dim0[15:0] in bits[31:16]
    g1[2] = NT1 << 16;                      // tensor_dim0[31:16]=0 | tensor_dim1 lo16
    g1[3] = (CH_ * 64) << 16;               // tensor_dim1 hi16=0 | tile_dim0
    g1[4] = NT1;                            // tile_dim1=8, tile_dim2=0
    g1[5] = K1STEPS * 64;                   // tensor_dim0_stride = 20480 elements
    g1[6] = 0;
    g1[7] = 0;
    v4i z4 = {};
#if __clang_major__ >= 23
    v8i z8 = {};
    __builtin_amdgcn_tensor_load_to_lds(g0, g1, z4, z4, z8, 0);
#else
    __builtin_amdgcn_tensor_load_to_lds(g0, g1, z4, z4, 0);
#endif
}
#endif

// ---------------------------------------------------------------------------
// Pack W1 (128x1280) and W2 (65x128, zero-padded to 80x128) into WMMA
// B-fragment order: frag f = ntile*KSTEPS + kstep; lane L holds
// B[k = kstep*4 + 2*(L/16) + j][n = ntile*16 + L%16] at packed[f*64 + L*2 + j].
// ---------------------------------------------------------------------------
__global__ __launch_bounds__(256) void ssd_pack_weights(
    const float* __restrict__ W1, const float* __restrict__ W2,
    float* __restrict__ packW1, float* __restrict__ packW2) {
    int t = blockIdx.x * 256 + threadIdx.x;
    if (t < NT1 * K1STEPS * 32) {
        int f = t >> 5, L = t & 31;
        int nt = f / K1STEPS, ks = f - nt * K1STEPS;
        int n = nt * 16 + (L & 15);
        int k = ks * 4 + 2 * (L >> 4);
        float2 v = *(const float2*)(W1 + n * CIN_ + k);
        *(float2*)(packW1 + f * 64 + L * 2) = v;
    } else if (t < NT1 * K1STEPS * 32 + NT2 * K2STEPS * 32) {
        int t2 = t - NT1 * K1STEPS * 32;
        int f = t2 >> 5, L = t2 & 31;
        int nt = f / K2STEPS, ks = f - nt * K2STEPS;
        int n = nt * 16 + (L & 15);
        int k = ks * 4 + 2 * (L >> 4);
        float2 v;
        if (n < OUTD_) v = *(const float2*)(W2 + n * HID_ + k);
        else { v.x = 0.0f; v.y = 0.0f; }
        *(float2*)(packW2 + f * 64 + L * 2) = v;
    }
}

// ---------------------------------------------------------------------------
// Fused head: TDM-staged W1 -> GEMM1(f32 WMMA) + bias + leakyReLU ->
// LDS shuffle -> GEMM2(f32 WMMA) + bias + activation epilogue.
// 8 waves/block; each wave owns one 16-row M tile across the full N.
// ---------------------------------------------------------------------------
__global__ __launch_bounds__(256) void ssd_fused_mlp(
    const float* __restrict__ feat,
    const float* __restrict__ packW1, const float* __restrict__ packW2,
    const float* __restrict__ b1, const float* __restrict__ b2,
    float* __restrict__ out) {
    // 64 KB: two 32 KB W1-chunk buffers during GEMM1, reused as the
    // 8-wave h-tile shuffle buffer (128 c x 16 m per wave) for GEMM2.
    __shared__ float smem[2 * CHUNK_FLOATS];

    const int lane = threadIdx.x & 31;
    const int wave = threadIdx.x >> 5;
    const int grp  = lane >> 4;     // lane group: selects K pair / upper M half
    const int lm   = lane & 15;     // M (A rows) or N (B, C cols)
    const int m0   = (blockIdx.x * 8 + wave) * 16;

    // Per-lane A-stream pointer: row m0+lm of features, K pair 2*grp.
    const int  mrowA = m0 + lm;
    const int  bA    = mrowA / HW_;
    const int  hwA   = mrowA - bA * HW_;
    const float* aptr = feat + (size_t)bA * (CIN_ * HW_) + hwA + (2 * grp) * HW_;

    v8f acc[NT1] = {};

#if HAVE_TDM
    if (wave == 0) {                       // uniform branch: only wave 0 runs TDM
        tdm_load_w1_chunk(packW1, 0, 0u);
        __builtin_amdgcn_s_wait_tensorcnt(0);
    }
    __syncthreads();
#endif

    for (int c = 0; c < NCHUNK; ++c) {
#if HAVE_TDM
        if (wave == 0 && (c + 1) < NCHUNK)     // async: overlap with compute below
            tdm_load_w1_chunk(packW1, c + 1, (unsigned)(((c + 1) & 1) * CHUNK_BYTES));
        const float* bbuf = smem + (c & 1) * CHUNK_FLOATS + lane * 2;
#else
        __syncthreads();
        for (int j4 = threadIdx.x; j4 < CHUNK_FLOATS / 4; j4 += 256) {
            int nt = j4 >> 8, off4 = j4 & 255;
            *(float4*)(smem + j4 * 4) =
                *(const float4*)(packW1 + nt * (K1STEPS * 64) + c * (CH_ * 64) + off4 * 4);
        }
        __syncthreads();
        const float* bbuf = smem + lane * 2;
#endif
        __builtin_prefetch(aptr + CH_ * 4 * HW_, 0, 0);   // A stream, one chunk ahead

        for (int ksl = 0; ksl < CH_; ++ksl) {
            v2f afrag;
            afrag.x = aptr[0];          // k = 4*ks + 2*grp
            afrag.y = aptr[HW_];        // k = 4*ks + 2*grp + 1
            aptr += 4 * HW_;
#pragma unroll
            for (int nt = 0; nt < NT1; ++nt) {
                float2 bv = *(const float2*)(bbuf + (nt * CH_ + ksl) * 64);
                v2f bfrag; bfrag.x = bv.x; bfrag.y = bv.y;
                acc[nt] = __builtin_amdgcn_wmma_f32_16x16x4_f32(
                    false, afrag, false, bfrag, (short)0, acc[nt], false, false);
            }
        }
#if HAVE_TDM
        if (wave == 0) __builtin_amdgcn_s_wait_tensorcnt(0);
        __syncthreads();
#endif
    }
    __syncthreads();   // chunk buffers dead; smem becomes the h shuffle buffer

    // ---- bias + leakyReLU, spill h tile to LDS (c-major: hl[c*16 + m]) ----
    float* hl = smem + wave * (HID_ * 16);
#pragma unroll
    for (int nt = 0; nt < NT1; ++nt) {
        const int c = nt * 16 + lm;
        const float bias = b1[c];
#pragma unroll
        for (int v = 0; v < 8; ++v) {
            float x = acc[nt][v] + bias;
            x = x > 0.0f ? x : 0.01f * x;
            hl[c * 16 + (v + 8 * grp)] = x;
        }
    }
    __syncthreads();

    // ---- GEMM2: out tile 16 x 80 (65 valid) over K=128 ----
    const float* b2base = packW2 + lane * 2;
    v8f acc2[NT2] = {};
    for (int ks = 0; ks < K2STEPS; ++ks) {
        const int c0 = ks * 4 + 2 * grp;
        v2f afrag;
        afrag.x = hl[c0 * 16 + lm];
        afrag.y = hl[(c0 + 1) * 16 + lm];
#pragma unroll
        for (int nt = 0; nt < NT2; ++nt) {
            float2 bv = *(const float2*)(b2base + (nt * K2STEPS + ks) * 64);
            v2f bfrag; bfrag.x = bv.x; bfrag.y = bv.y;
            acc2[nt] = __builtin_amdgcn_wmma_f32_16x16x4_f32(
                false, afrag, false, bfrag, (short)0, acc2[nt], false, false);
        }
    }

    // ---- epilogue: per-channel activation + scatter to reference layout ----
#pragma unroll
    for (int nt = 0; nt < NT2; ++nt) {
        const int o = nt * 16 + lm;
        const float bias = (o < OUTD_) ? b2[o] : 0.0f;
#pragma unroll
        for (int v = 0; v < 8; ++v) {
            const int m  = m0 + v + 8 * grp;
            const int bb = m / HW_;
            const int hw = m - bb * HW_;
            const float x = acc2[nt][v] + bias;
            if (o < 36) {                        // offsets (B, 9, 4, 49)
                const int a = o >> 2, comp = o & 3;
                const float val = (comp < 2) ? (sigmoid_f(x) - 0.5f) : x;
                out[OFF_OFFS + bb * (A_ * 4 * HW_) + a * (4 * HW_) + comp * HW_ + hw] = val;
            } else if (o < 45) {                 // conf (B, 9, 49)
                out[OFF_CONF + bb * (A_ * HW_) + (o - 36) * HW_ + hw] = sigmoid_f(x);
            } else if (o < OUTD_) {              // class scores (B, 20, 49)
                out[OFF_CLS + bb * (NCLS_ * HW_) + (o - 45) * HW_ + hw] = x;
            }
        }
    }
}

// ---------------------------------------------------------------------------
// IoU matrix: one block per batch; GT boxes staged in LDS; 441 anchors x 40 GT.
// ---------------------------------------------------------------------------
__global__ __launch_bounds__(256) void ssd_iou(
    const float* __restrict__ grid, const float* __restrict__ bboxes,
    const float* __restrict__ anc, float* __restrict__ out_iou) {
    const int b = blockIdx.x;
    __shared__ float4 sbox[NGT_];
    __shared__ float  sarea[NGT_];
    __shared__ float  sanc[2 * A_];
    const int tid = threadIdx.x;
    if (tid < NGT_) {
        const float* bp = bboxes + (size_t)(b * NGT_ + tid) * 5;
        float4 v; v.x = bp[0]; v.y = bp[1]; v.z = bp[2]; v.w = bp[3];
        sbox[tid]  = v;
        sarea[tid] = (v.z - v.x) * (v.w - v.y);
    }
    if (tid < 2 * A_) sanc[tid] = anc[tid];
    __syncthreads();

    for (int p = tid; p < A_ * HW_; p += 256) {
        const int a  = p / HW_;
        const int hw = p - a * HW_;
        const float cx = grid[(b * HW_ + hw) * 2 + 0];
        const float cy = grid[(b * HW_ + hw) * 2 + 1];
        const float hx = sanc[a * 2 + 0] * 0.5f;
        const float hy = sanc[a * 2 + 1] * 0.5f;
        const float px1 = cx - hx, py1 = cy - hy, px2 = cx + hx, py2 = cy + hy;
        const float pmin  = fminf(fminf(px1, py1), fminf(px2, py2));
        const float legal = pmin > 0.0f ? 1.0f : 0.0f;
        const float parea = (px2 - px1) * (py2 - py1);

        float vals[NGT_];
#pragma unroll
        for (int g = 0; g < NGT_; ++g) {
            const float4 gb = sbox[g];
            const float iw = fmaxf(fminf(px2, gb.z) - fmaxf(px1, gb.x), 0.0f);
            const float ih = fmaxf(fminf(py2, gb.w) - fmaxf(py1, gb.y), 0.0f);
            const float inter = iw * ih * legal;
            vals[g] = inter / (sarea[g] + parea - inter);
        }
        float* op = out_iou + (size_t)(b * (A_ * HW_) + p) * NGT_;
#pragma unroll
        for (int i = 0; i < NGT_ / 4; ++i)
            *(float4*)(op + i * 4) = make_float4(vals[i*4], vals[i*4+1], vals[i*4+2], vals[i*4+3]);
    }
}

extern "C" void kernel_launch(void* const* d_in, const int* in_sizes, int n_in,
                              void* d_out, int out_size, void* d_ws, size_t ws_size,
                              hipStream_t stream) {
    const float* feat   = (const float*)d_in[0];
    const float* grid   = (const float*)d_in[1];
    const float* bboxes = (const float*)d_in[2];
    const float* anc    = (const float*)d_in[3];
    const float* W1     = (const float*)d_in[4];
    const float* b1     = (const float*)d_in[5];
    const float* W2     = (const float*)d_in[6];
    const float* b2     = (const float*)d_in[7];
    float* out = (float*)d_out;

    float* packW1 = (float*)d_ws;
    float* packW2 = packW1 + PACKW1_FLOATS;

    // 1) pack weights into WMMA B-fragment order (deterministic, every call)
    const int pack_threads = NT1 * K1STEPS * 32 + NT2 * K2STEPS * 32; // 87040
    ssd_pack_weights<<<(pack_threads + 255) / 256, 256, 0, stream>>>(W1, W2, packW1, packW2);

    // 2) fused MLP head: 784 M-tiles, 8 waves/block -> 98 blocks
    ssd_fused_mlp<<<M_TOT / (16 * 8), 256, 0, stream>>>(feat, packW1, packW2, b1, b2, out);

    // 3) IoU matrix
    ssd_iou<<<B_, 256, 0, stream>>>(grid, bboxes, anc, out + OFF_IOU);
}